// FGSAttn_76613626626129
// MI455X (gfx1250) — compile-verified
//
#include <hip/hip_runtime.h>

typedef float v2f __attribute__((ext_vector_type(2)));
typedef float v8f __attribute__((ext_vector_type(8)));

#define NB   32      // batch
#define NG   4       // groups
#define NK   64      // channels per group
#define NC   256     // channels
#define HH   56      // H == W
#define HW   (HH*HH) // 3136
#define PAD  64      // padded DFT size
#define NP   (PAD*PAD) // 4096
#define NBIN 29      // radial bins (min(28,28)/d + 1, d=1)
#define NIMG (NB*NG) // 128
#define TPB  512     // threads per image block (16 waves)

// ---------------------------------------------------------------------------
// Kernel A: comp = mean_k + max_k over the K=64 channels of each group,
// written into a zero-padded 64x64 tile per image. Memory-bound: 103 MB read.
// ---------------------------------------------------------------------------
__global__ void comp_kernel(const float* __restrict__ feature,
                            float* __restrict__ comp) {
  int idx = blockIdx.x * blockDim.x + threadIdx.x;   // < NIMG*NP = 524288
  int img = idx >> 12;                                // /4096
  int p   = idx & (NP - 1);
  int u = p >> 6, w = p & 63;
  float out = 0.f;
  if (u < HH && w < HH) {
    const float* base = feature + (size_t)img * NK * HW + u * HH + w;
    float s = 0.f, mx = -__builtin_inff();
    #pragma unroll 8
    for (int k = 0; k < NK; ++k) {
      float v = base[(size_t)k * HW];
      s += v;
      mx = fmaxf(mx, v);
    }
    out = s * (1.f / (float)NK) + mx;
  }
  comp[idx] = out;
}

// ---------------------------------------------------------------------------
// WMMA helpers: 64x64x64 f32 GEMM tile via V_WMMA_F32_16X16X4_F32.
// A fragment (16x4): lane l holds row m=l&15, K = kk + 2*(l>>4) + j  (vgpr j)
// B fragment (4x16): lane l holds col n=l&15, K = kk + 2*(l>>4) + j  (vgpr j)
// C/D (16x16): vgpr r, lanes 0-15 -> M = r, lanes 16-31 -> M = r+8
// ---------------------------------------------------------------------------
__device__ __forceinline__ v8f mm64(const float* __restrict__ A,
                                    const float* __restrict__ Bm,
                                    int mi, int ni, int lane, v8f acc) {
  const int m  = mi * 16 + (lane & 15);
  const int n  = ni * 16 + (lane & 15);
  const int kh = (lane >> 4) << 1;
  #pragma unroll
  for (int kk = 0; kk < PAD; kk += 4) {
    v2f a, b;
    a.x = A[m * PAD + kk + kh + 0];
    a.y = A[m * PAD + kk + kh + 1];
    b.x = Bm[(kk + kh + 0) * PAD + n];
    b.y = Bm[(kk + kh + 1) * PAD + n];
    acc = __builtin_amdgcn_wmma_f32_16x16x4_f32(
        false, a, false, b, (short)0, acc, false, false);
  }
  return acc;
}

__device__ __forceinline__ void store_tile(float* __restrict__ OUT,
                                           int mi, int ni, int lane, v8f acc) {
  const int n  = ni * 16 + (lane & 15);
  const int mb = mi * 16 + ((lane >> 4) << 3);
  #pragma unroll
  for (int r = 0; r < 8; ++r) OUT[(mb + r) * PAD + n] = acc[r];
}

__device__ __forceinline__ int radial_label(int u, int h) {
  // fftshift-ed coordinates of spectral pixel (u,h); +28 is self-inverse mod 56
  int i2 = ((u + 28) % HH) - 28;
  int j2 = ((h + 28) % HH) - 28;
  float r = sqrtf((float)(i2 * i2 + j2 * j2));
  return (int)fminf(floorf(r), (float)(NBIN - 1));   // d = 1 -> ceil(rf) == rf
}

// ---------------------------------------------------------------------------
// Kernel B (fused spectral pipeline, one workgroup per (b,g) image):
//   DFT fwd: T = X@F ; S = F@T        (F symmetric, F = Fr + i*Fi)
//   radial-bin pooled |S|, 29x29 FC + leaky-relu  -> att[29]
//   M = S * att[label]                (exp(i*phase)*amp*att == S*att)
//   DFT inv: U = M@conj(F) ; nf = Re(conj(F)@U)/3136
//   min-max normalize nf over 56x56 -> attn_out
// ---------------------------------------------------------------------------
__global__ void __launch_bounds__(TPB)
fgs_fft_kernel(const float* __restrict__ comp,
               const float* __restrict__ fc_w,
               const float* __restrict__ fc_b,
               float* __restrict__ attn_out) {
  extern __shared__ float smem[];
  float* FR  = smem;            // 4096  Fr
  float* FI  = FR  + NP;        // 4096  Fi
  float* B0  = FI  + NP;        // 4096  X -> Sr(=Mr) -> nf
  float* B1  = B0  + NP;        // 4096  Si(=Mi)
  float* B2  = B1  + NP;        // 4096  Tr -> Ur
  float* B3  = B2  + NP;        // 4096  Ti -> Ui
  float* red  = B3  + NP;       // 512   reduction scratch
  float* binS = red + TPB;      // 32
  float* binC = binS + 32;      // 32
  float* attA = binC + 32;      // 32

  const int tid  = threadIdx.x;
  const int lane = tid & 31;
  const int wid  = tid >> 5;          // 0..15
  const int mi   = wid >> 2, ni = wid & 3;
  const int img  = blockIdx.x;        // b*NG + g
  const int g    = img & (NG - 1);

  // ---- fill DFT matrices (zero padded) + load comp tile --------------------
  for (int p = tid; p < NP; p += TPB) {
    int u = p >> 6, h = p & 63;
    float fr = 0.f, fi = 0.f;
    if (u < HH && h < HH) {
      int m = (u * h) % HH;
      float ang = -6.283185307179586f * (float)m * (1.f / (float)HH);
      fr = __cosf(ang);
      fi = __sinf(ang);
    }
    FR[p] = fr; FI[p] = fi;
    B0[p] = comp[(size_t)img * NP + p];
  }
  if (tid < 32) { binS[tid] = 0.f; binC[tid] = 0.f; }
  __syncthreads();

  const v8f z = {};

  // ---- stage 1: Tr = X@Fr -> B2 ; Ti = X@Fi -> B3 --------------------------
  v8f tr = mm64(B0, FR, mi, ni, lane, z);
  v8f ti = mm64(B0, FI, mi, ni, lane, z);
  store_tile(B2, mi, ni, lane, tr);
  store_tile(B3, mi, ni, lane, ti);
  __syncthreads();

  // ---- stage 2: Sr = Fr@Tr - Fi@Ti -> B0 ; Si = Fr@Ti + Fi@Tr -> B1 --------
  v8f a1 = mm64(FR, B2, mi, ni, lane, z);
  v8f a2 = mm64(FI, B3, mi, ni, lane, z);
  v8f sr = a1 - a2;
  v8f si = mm64(FR, B3, mi, ni, lane, z);
  si     = mm64(FI, B2, mi, ni, lane, si);
  store_tile(B0, mi, ni, lane, sr);
  store_tile(B1, mi, ni, lane, si);
  __syncthreads();

  // ---- radial-bin pooling of amplitude ------------------------------------
  for (int p = tid; p < NP; p += TPB) {
    int u = p >> 6, h = p & 63;
    if (u < HH && h < HH) {
      int lab = radial_label(u, h);
      float amp = sqrtf(B0[p] * B0[p] + B1[p] * B1[p]);
      atomicAdd(&binS[lab], amp);   // ds_add_f32
      atomicAdd(&binC[lab], 1.f);
    }
  }
  __syncthreads();

  // ---- 29x29 FC + leaky relu ----------------------------------------------
  if (tid < NBIN) {
    float s = fc_b[g * NBIN + tid];
    #pragma unroll
    for (int n = 0; n < NBIN; ++n)
      s += (binS[n] / binC[n]) * fc_w[(g * NBIN + tid) * NBIN + n];
    attA[tid] = (s > 0.f) ? s : 0.01f * s;
  }
  __syncthreads();

  // ---- scale spectrum by attention ----------------------------------------
  for (int p = tid; p < NP; p += TPB) {
    int u = p >> 6, h = p & 63;
    if (u < HH && h < HH) {
      float s = attA[radial_label(u, h)];
      B0[p] *= s;
      B1[p] *= s;
    }
  }
  __syncthreads();

  // ---- stage 3: Ur = Mr@Fr + Mi@Fi -> B2 ; Ui = Mi@Fr - Mr@Fi -> B3 --------
  v8f ur = mm64(B0, FR, mi, ni, lane, z);
  ur     = mm64(B1, FI, mi, ni, lane, ur);
  v8f u1 = mm64(B1, FR, mi, ni, lane, z);
  v8f u2 = mm64(B0, FI, mi, ni, lane, z);
  v8f ui = u1 - u2;
  store_tile(B2, mi, ni, lane, ur);
  store_tile(B3, mi, ni, lane, ui);
  __syncthreads();

  // ---- stage 4: nf = (Fr@Ur + Fi@Ui)/3136 -> B0 ----------------------------
  v8f nf = mm64(FR, B2, mi, ni, lane, z);
  nf     = mm64(FI, B3, mi, ni, lane, nf);
  nf     = nf * (1.f / (float)HW);
  store_tile(B0, mi, ni, lane, nf);
  __syncthreads();

  // ---- min/max over 56x56 and normalized write-out -------------------------
  float mn = __builtin_inff(), mx = -__builtin_inff();
  for (int p = tid; p < NP; p += TPB) {
    int u = p >> 6, h = p & 63;
    if (u < HH && h < HH) {
      float v = B0[p];
      mn = fminf(mn, v);
      mx = fmaxf(mx, v);
    }
  }
  red[tid] = mn;
  __syncthreads();
  for (int s = TPB / 2; s > 0; s >>= 1) {
    if (tid < s) red[tid] = fminf(red[tid], red[tid + s]);
    __syncthreads();
  }
  float gmn = red[0];
  __syncthreads();
  red[tid] = mx;
  __syncthreads();
  for (int s = TPB / 2; s > 0; s >>= 1) {
    if (tid < s) red[tid] = fmaxf(red[tid], red[tid + s]);
    __syncthreads();
  }
  float gmx = red[0];

  float inv = 1.f / (gmx - gmn);
  for (int p = tid; p < NP; p += TPB) {
    int u = p >> 6, h = p & 63;
    if (u < HH && h < HH)
      attn_out[(size_t)img * HW + u * HH + h] = (B0[p] - gmn) * inv;
  }
}

// ---------------------------------------------------------------------------
// Kernel E: out = feature * (1 + gamma[c] * attn[b,g,h,w]), float4 vectorized.
// Memory-bound: ~207 MB of traffic.
// ---------------------------------------------------------------------------
__global__ void apply_kernel(const float4* __restrict__ f4,
                             const float* __restrict__ gamma,
                             const float* __restrict__ attn,
                             float4* __restrict__ out4) {
  int i = blockIdx.x * blockDim.x + threadIdx.x;  // < 6422528
  int flat = i << 2;
  int hw = flat % HW;                 // flat%4==0 and HW%4==0 -> hw%4==0
  int c  = (flat / HW) & (NC - 1);
  int b  = flat / (HW * NC);
  int g  = c >> 6;
  float ga = gamma[c];
  float4 fv = f4[i];
  const float4 av = *(const float4*)(attn + ((size_t)(b * NG + g) * HW + hw));
  float4 o;
  o.x = fv.x * (1.f + ga * av.x);
  o.y = fv.y * (1.f + ga * av.y);
  o.z = fv.z * (1.f + ga * av.z);
  o.w = fv.w * (1.f + ga * av.w);
  out4[i] = o;
}

// ---------------------------------------------------------------------------
extern "C" void kernel_launch(void* const* d_in, const int* in_sizes, int n_in,
                              void* d_out, int out_size, void* d_ws, size_t ws_size,
                              hipStream_t stream) {
  const float* feature = (const float*)d_in[0];  // (32,256,56,56)
  const float* gamma   = (const float*)d_in[1];  // (256,)
  const float* fc_w    = (const float*)d_in[2];  // (4,29,29)
  const float* fc_b    = (const float*)d_in[3];  // (4,29)

  float* comp = (float*)d_ws;                    // NIMG*NP   = 524288 floats
  float* attn = comp + (size_t)NIMG * NP;        // NIMG*HW   = 401408 floats

  // A: group reduce into padded tiles (524288 elements, 256 thr/blk)
  comp_kernel<<<(NIMG * NP) / 256, 256, 0, stream>>>(feature, comp);

  // B: fused DFT -> pool -> FC -> scale -> iDFT -> normalize (one block/image)
  size_t shmem = (size_t)(6 * NP + TPB + 96) * sizeof(float);  // ~100.7 KB
  fgs_fft_kernel<<<NIMG, TPB, shmem, stream>>>(comp, fc_w, fc_b, attn);

  // E: elementwise apply (float4)
  int n4 = (NB * NC * HW) / 4;                   // 6422528
  apply_kernel<<<n4 / 256, 256, 0, stream>>>((const float4*)feature, gamma,
                                             attn, (float4*)d_out);
}